// MultiHeadSelfAttention_74156905333486
// MI455X (gfx1250) — compile-verified
//
#include <hip/hip_runtime.h>
#include <hip/hip_bf16.h>

// Problem constants (match reference)
#define BB 4
#define TT 2048
#define DD 768
#define HH 12
#define HD 64

typedef float v2f __attribute__((ext_vector_type(2)));
typedef float v8f __attribute__((ext_vector_type(8)));
typedef int   v4i __attribute__((ext_vector_type(4)));
typedef int   v8i __attribute__((ext_vector_type(8)));
typedef unsigned int u32;
typedef unsigned long long u64;

// D = A(16x4 f32) * B(4x16 f32) + C(16x16 f32)
__device__ __forceinline__ v8f wmma_f32(v2f a, v2f b, v8f c) {
    return __builtin_amdgcn_wmma_f32_16x16x4_f32(false, a, false, b, (short)0, c, false, false);
}

// ---------------------------------------------------------------------------
// Tensor Data Mover: 2D f32 tile (tile_h x tile_w, row stride stride_elems)
// from global -> LDS byte offset lds_off (rows packed, tile_w floats each).
// D# built in SGPRs per CDNA5 ISA 8.3/8.4; issued via inline asm (portable
// across both toolchains' builtin arities). Tracked by TENSORcnt.
// ---------------------------------------------------------------------------
__device__ __forceinline__ void tdm_load_2d_f32(u32 lds_off, const float* gsrc,
                                                u32 tile_w, u32 tile_h,
                                                u32 stride_elems)
{
    const u64 ga = (u64)(uintptr_t)gsrc;
    v4i g0;
    g0[0] = 1;                                            // count=1 (valid user D#)
    g0[1] = (int)lds_off;                                 // lds_addr (bytes)
    g0[2] = (int)(u32)ga;                                 // global_addr[31:0]
    g0[3] = (int)((u32)((ga >> 32) & 0x01FFFFFFu) | (2u << 30)); // addr[56:32] | type=2
    v8i g1;
    g1[0] = (int)(2u << 16);                              // data_size = 4 bytes
    g1[1] = (int)((tile_w & 0xFFFFu) << 16);              // tensor_dim0[15:0]
    g1[2] = (int)((tile_w >> 16) | ((tile_h & 0xFFFFu) << 16)); // dim0 hi | dim1 lo
    g1[3] = (int)((tile_h >> 16) | ((tile_w & 0xFFFFu) << 16)); // dim1 hi | tile_dim0
    g1[4] = (int)(tile_h & 0xFFFFu);                      // tile_dim1 (tile_dim2=0)
    g1[5] = (int)stride_elems;                            // tensor_dim0_stride[31:0]
    g1[6] = 0;                                            // stride hi | dim1_stride lo
    g1[7] = 0;
    asm volatile("tensor_load_to_lds %0, %1" :: "s"(g0), "s"(g1) : "memory");
}

// ---------------------------------------------------------------------------
// C[M,N] = X[M,768] * W[N,768]^T (row-major). Block: 8 waves, 32x128 tile.
// X tile (32x768 = 96KB) staged by one TDM op; each wave computes TWO M-tiles
// for its 16-column strip, so each global B-fragment feeds 2 WMMAs. W streams
// from global (w_qkv 7MB lives in the 192MB L2 across all row-tile re-reads).
// ---------------------------------------------------------------------------
__global__ __launch_bounds__(256)
void gemm_xwt_kernel(const float* __restrict__ X, const float* __restrict__ W,
                     float* __restrict__ C, int N)
{
    extern __shared__ float xs[];   // 32*768 floats = 96 KB of 320 KB/WGP

    const int row0 = blockIdx.x * 32;
    const int col0 = blockIdx.y * 128;
    const int tid  = threadIdx.x;

    // Stage X tile with the Tensor Data Mover (wave 0 issues; EXEC ignored)
    if (tid < 32) {
        tdm_load_2d_f32((u32)(uintptr_t)xs, X + (size_t)row0 * DD, DD, 32, DD);
        __builtin_amdgcn_s_wait_tensorcnt(0);
    }
    __syncthreads();

    const int wave = tid >> 5;
    const int lane = tid & 31;
    const int half = lane >> 4;     // which K-pair within the 4-deep slice
    const int l16  = lane & 15;

    const int ncol = col0 + wave * 16 + l16;       // this lane's W row (= C col)
    const float* __restrict__ wrow  = W + (size_t)ncol * DD;
    const float* __restrict__ xrow0 = &xs[l16 * DD + half * 2];          // M-tile 0
    const float* __restrict__ xrow1 = &xs[(16 + l16) * DD + half * 2];   // M-tile 1

    v8f acc0 = {}, acc1 = {};
    #pragma unroll 4
    for (int k0 = 0; k0 < DD; k0 += 4) {
        if ((k0 & 15) == 0)
            __builtin_prefetch(wrow + k0 + 256, 0, 1);   // global_prefetch_b8
        v2f a0, a1, b;
        b.x  = wrow[k0 + half * 2];       // global_load_b64: B 4x16, N=l16
        b.y  = wrow[k0 + half * 2 + 1];
        a0.x = xrow0[k0];                 // ds_load_b64: A 16x4, M=l16, K=half*2+{0,1}
        a0.y = xrow0[k0 + 1];
        a1.x = xrow1[k0];
        a1.y = xrow1[k0 + 1];
        acc0 = wmma_f32(a0, b, acc0);     // B fragment reused by both M-tiles
        acc1 = wmma_f32(a1, b, acc1);
    }

    // C layout: VGPR r holds (M = r + half*8, N = l16)
    #pragma unroll
    for (int r = 0; r < 8; ++r) {
        C[(size_t)(row0 + r + half * 8) * N + ncol]      = acc0[r];
        C[(size_t)(row0 + 16 + r + half * 8) * N + ncol] = acc1[r];
    }
}

// ---------------------------------------------------------------------------
// In-place RoPE on the q and k slices of the fused qkv buffer (B,T,3*768).
// ---------------------------------------------------------------------------
__global__ __launch_bounds__(256)
void rope_kernel(float* __restrict__ qkv, const float* __restrict__ cosw,
                 const float* __restrict__ sinw)
{
    const int PAIRS = BB * TT * HH * (HD / 2);     // per q or k
    int idx = blockIdx.x * blockDim.x + threadIdx.x;
    if (idx >= 2 * PAIRS) return;
    const int which = idx / PAIRS;                 // 0 = q, 1 = k
    int r = idx - which * PAIRS;
    const int i  = r & 31;
    int u  = r >> 5;
    const int h  = u % HH;
    const int bt = u / HH;                         // b*T + t
    const int t  = bt & (TT - 1);

    float* p = qkv + (size_t)bt * (3 * DD) + which * DD + h * HD;
    const float c = cosw[t * 32 + i];
    const float s = sinw[t * 32 + i];
    const float x1 = p[i];
    const float x2 = p[i + 32];
    p[i]      = x1 * c - x2 * s;
    p[i + 32] = x2 * c + x1 * s;
}

// ---------------------------------------------------------------------------
// Flash attention. One wave owns a 16-query tile of one (b,h).
// Double-buffered TDM pipeline: while tile j is consumed by WMMAs, the TDM
// streams tile j+1 into the other LDS buffer; s_wait_tensorcnt(2) exploits
// in-order TENSORcnt completion to wait only for tile j's pair of loads.
// Block = 8 waves; dynamic LDS: 8 * (2*1024 K + 2*1024 V + 256 P) = 136 KB.
// ---------------------------------------------------------------------------
__global__ __launch_bounds__(256)
void attn_kernel(const float* __restrict__ qkv, const float* __restrict__ bias,
                 float* __restrict__ outp)
{
    extern __shared__ float smem[];

    const int tid  = threadIdx.x;
    const int lane = tid & 31;
    const int half = lane >> 4;
    const int l16  = lane & 15;
    // Scalar wave id (first lane's tid >> 5) so TDM descriptors stay in SGPRs
    const int wv   = __builtin_amdgcn_readfirstlane((int)tid) >> 5;

    float* slab  = smem + wv * 4352;
    float* kbufs[2] = { slab,        slab + 1024 };     // 16x64 K tiles
    float* vbufs[2] = { slab + 2048, slab + 3072 };     // 16x64 V tiles
    float* P        = slab + 4096;                      // 16x16 P tile
    const u32 k_lds[2] = { (u32)(uintptr_t)kbufs[0], (u32)(uintptr_t)kbufs[1] };
    const u32 v_lds[2] = { (u32)(uintptr_t)vbufs[0], (u32)(uintptr_t)vbufs[1] };

    const int gw = blockIdx.x * 8 + wv;            // 0 .. B*H*128-1 (scalar)
    const int qt = gw & 127;                       // query tile
    const int h  = (gw >> 7) % HH;
    const int b  = (gw >> 7) / HH;
    const int q0 = qt * 16;

    const size_t rs = 3 * DD;                      // qkv row stride
    const float* __restrict__ qbase = qkv + (size_t)(b * TT) * rs + h * HD;
    const float* __restrict__ kbase = qbase + DD;
    const float* __restrict__ vbase = qbase + 2 * DD;
    const float* __restrict__ brow  = bias + (size_t)b * TT * TT;

    // Prologue: stream key-tile 0 into buffer 0
    tdm_load_2d_f32(k_lds[0], kbase, HD, 16, (u32)rs);
    tdm_load_2d_f32(v_lds[0], vbase, HD, 16, (u32)rs);

    // Q A-fragments for all 16 K-steps (HD=64): lane holds M=l16, K-pair by half
    v2f qf[16];
    #pragma unroll
    for (int kk = 0; kk < 16; ++kk) {
        const float* qp = qbase + (size_t)(q0 + l16) * rs + kk * 4 + half * 2;
        qf[kk].x = qp[0];
        qf[kk].y = qp[1];
    }

    float m_run[8], l_run[8];
    #pragma unroll
    for (int r = 0; r < 8; ++r) { m_run[r] = -3.0e38f; l_run[r] = 0.0f; }
    v8f o0 = {}, o1 = {}, o2 = {}, o3 = {};
    v8f* o[4] = { &o0, &o1, &o2, &o3 };

    const float scale = 0.125f;                    // 1/sqrt(64)

    for (int j = 0; j < TT / 16; ++j) {
        const int cur = j & 1;
        const int k0r = j * 16;

        // Issue next tile's TDM pair behind this tile's compute, then wait
        // only for the current tile (in-order TENSORcnt: <=2 outstanding).
        if (j + 1 < TT / 16) {
            const int nxt = cur ^ 1;
            tdm_load_2d_f32(k_lds[nxt], kbase + (size_t)(k0r + 16) * rs, HD, 16, (u32)rs);
            tdm_load_2d_f32(v_lds[nxt], vbase + (size_t)(k0r + 16) * rs, HD, 16, (u32)rs);
            __builtin_amdgcn_s_wait_tensorcnt(2);
        } else {
            __builtin_amdgcn_s_wait_tensorcnt(0);
        }
        const float* kbuf = kbufs[cur];
        const float* vbuf = vbufs[cur];

        // S = Q * K^T : 16 WMMAs; B[kd][n] = K[n][kd] read from LDS
        v8f s = {};
        #pragma unroll
        for (int kk = 0; kk < 16; ++kk) {
            v2f bf;
            bf.x = kbuf[l16 * HD + kk * 4 + half * 2];
            bf.y = kbuf[l16 * HD + kk * 4 + half * 2 + 1];
            s = wmma_f32(qf[kk], bf, s);
        }

        // scale + bias + online softmax (row m lives in one VGPR across a 16-lane half)
        #pragma unroll
        for (int r = 0; r < 8; ++r) {
            const int qm = q0 + r + half * 8;
            float sv = s[r] * scale + brow[(size_t)qm * TT + k0r + l16];
            float mx = sv;
            mx = fmaxf(mx, __shfl_xor(mx, 1, 16));
            mx = fmaxf(mx, __shfl_xor(mx, 2, 16));
            mx = fmaxf(mx, __shfl_xor(mx, 4, 16));
            mx = fmaxf(mx, __shfl_xor(mx, 8, 16));
            const float m_new = fmaxf(m_run[r], mx);
            const float alpha = __expf(m_run[r] - m_new);
            const float p     = __expf(sv - m_new);
            float ps = p;
            ps += __shfl_xor(ps, 1, 16);
            ps += __shfl_xor(ps, 2, 16);
            ps += __shfl_xor(ps, 4, 16);
            ps += __shfl_xor(ps, 8, 16);
            l_run[r] = l_run[r] * alpha + ps;
            m_run[r] = m_new;
            (*o[0])[r] *= alpha;
            (*o[1])[r] *= alpha;
            (*o[2])[r] *= alpha;
            (*o[3])[r] *= alpha;
            P[(r + half * 8) * 16 + l16] = p;      // re-layout C -> A via LDS
        }
        // LDS ops are in-order within the wave; loads below depend via DScnt.

        // O += P * V : 4 K-steps x 4 N-tiles, V fragments from LDS
        #pragma unroll
        for (int kk = 0; kk < 4; ++kk) {
            v2f a;
            a.x = P[l16 * 16 + kk * 4 + half * 2];
            a.y = P[l16 * 16 + kk * 4 + half * 2 + 1];
            const int vrow = kk * 4 + half * 2;
            #pragma unroll
            for (int nt = 0; nt < 4; ++nt) {
                v2f bv;
                bv.x = vbuf[vrow * HD + nt * 16 + l16];
                bv.y = vbuf[(vrow + 1) * HD + nt * 16 + l16];
                *o[nt] = wmma_f32(a, bv, *o[nt]);
            }
        }
    }

    // Normalize and write (B,T,H*HD)
    #pragma unroll
    for (int r = 0; r < 8; ++r) {
        const float inv = 1.0f / l_run[r];
        const int t = q0 + r + half * 8;
        #pragma unroll
        for (int nt = 0; nt < 4; ++nt)
            outp[(size_t)(b * TT + t) * DD + h * HD + nt * 16 + l16] = (*o[nt])[r] * inv;
    }
}

// ---------------------------------------------------------------------------
extern "C" void kernel_launch(void* const* d_in, const int* in_sizes, int n_in,
                              void* d_out, int out_size, void* d_ws, size_t ws_size,
                              hipStream_t stream) {
    const float* x         = (const float*)d_in[0];
    const float* attn_bias = (const float*)d_in[1];
    const float* cosw      = (const float*)d_in[2];
    const float* sinw      = (const float*)d_in[3];
    const float* w_qkv     = (const float*)d_in[4];
    const float* w_proj    = (const float*)d_in[5];
    float* out = (float*)d_out;

    float* qkv      = (float*)d_ws;                        // B*T*3D floats (75.5 MB)
    float* attn_out = qkv + (size_t)BB * TT * 3 * DD;      // B*T*D  floats (25.2 MB)

    dim3 blk(256);
    const size_t gemm_lds = 32 * DD * sizeof(float);       // 96 KB
    const size_t attn_lds = 8 * 4352 * sizeof(float);      // 136 KB

    // 1) fused QKV projection: (8192 x 768) @ (2304 x 768)^T
    dim3 g1(BB * TT / 32, 3 * DD / 128);
    gemm_xwt_kernel<<<g1, blk, gemm_lds, stream>>>(x, w_qkv, qkv, 3 * DD);

    // 2) RoPE on q,k slices
    const int pairs = 2 * BB * TT * HH * (HD / 2);
    rope_kernel<<<(pairs + 255) / 256, blk, 0, stream>>>(qkv, cosw, sinw);

    // 3) flash attention (double-buffered TDM; 136 KB dynamic LDS)
    attn_kernel<<<BB * HH * (TT / 16) / 8, blk, attn_lds, stream>>>(
        qkv, attn_bias, attn_out);

    // 4) output projection: (8192 x 768) @ (768 x 768)^T
    dim3 g3(BB * TT / 32, DD / 128);
    gemm_xwt_kernel<<<g3, blk, gemm_lds, stream>>>(attn_out, w_proj, out, DD);
}